// MyModel_84026740179687
// MI455X (gfx1250) — compile-verified
//
#include <hip/hip_runtime.h>

// CDNA5 / gfx1250: wave32, WMMA f32 16x16x4 (K=3 padded to 4).
typedef __attribute__((ext_vector_type(2))) float v2f;
typedef __attribute__((ext_vector_type(8))) float v8f;

#define FDIM 256
#define NPT  20   // points per batch element
#define NTASK (NPT * (FDIM / 16))  // 320 (i, feature-tile) tasks per block

// Fast silu using native CDNA5 transcendentals:
//   silu(x) = x * sigmoid(x) = x * rcp(1 + exp2(-x * log2(e)))
// Lowers to v_mul / v_exp_f32 / v_add / v_rcp_f32 / v_mul (TRANS ops
// co-execute with VALU/WMMA), instead of the ~20-op accurate OCML expansion.
__device__ __forceinline__ float silu_f(float x) {
    const float LOG2E = 1.44269504088896340736f;
    float e = __builtin_amdgcn_exp2f(-x * LOG2E);
    return x * __builtin_amdgcn_rcpf(1.0f + e);
}

__global__ __launch_bounds__(256) void edge_mlp_kernel(
    const float* __restrict__ r,    // (B, 20, 3)
    const float* __restrict__ W1,   // (3, 256)
    const float* __restrict__ b1,   // (256)
    const float* __restrict__ W2,   // (3, 256)
    const float* __restrict__ b2,   // (256)
    float* __restrict__ partial)    // (B) per-block partial sums
{
    __shared__ float sr[NPT * 3];
    __shared__ float sW1[3 * FDIM];
    __shared__ float sW2[3 * FDIM];
    __shared__ float sb1[FDIM];
    __shared__ float sb2[FDIM];
    __shared__ float swave[8];

    const int b = blockIdx.x;
    const int t = threadIdx.x;

    // Stage positions + weights + biases in LDS (tiny, reused 320x per block).
    if (t < NPT * 3) sr[t] = r[b * (NPT * 3) + t];
    for (int k = t; k < 3 * FDIM; k += 256) {
        sW1[k] = W1[k];
        sW2[k] = W2[k];
    }
    sb1[t] = b1[t];   // blockDim.x == FDIM == 256
    sb2[t] = b2[t];
    __syncthreads();

    const int wave = t >> 5;
    const int lane = t & 31;
    const int hi   = lane >> 4;      // wave half
    const int lm   = lane & 15;      // A: row M; B/C/D: column N
    const int c0   = 2 * hi;         // K for A/B vgpr0 (0 or 2)
    const int c1   = 2 * hi + 1;     // K for A/B vgpr1 (1 or 3; 3 is zero-pad)

    float wsum = 0.0f;

    // 320 tasks / 8 waves = exactly 40 iterations per wave (uniform EXEC for WMMA).
    for (int task = wave; task < NTASK; task += 8) {
        const int i  = task >> 4;          // point index (row of h)
        const int ft = task & 15;          // feature tile
        const int n  = ft * 16 + lm;       // global feature column

        // B tiles (W slices), K = v + 2*hi, K==3 padded with 0.
        v2f B1t, B2t;
        B1t.x = sW1[c0 * FDIM + n];
        B1t.y = (c1 < 3) ? sW1[c1 * FDIM + n] : 0.0f;
        B2t.x = sW2[c0 * FDIM + n];
        B2t.y = (c1 < 3) ? sW2[c1 * FDIM + n] : 0.0f;

        const float bias1 = sb1[n];
        const float bias2 = sb2[n];

        const float ri0 = sr[i * 3 + 0];
        const float ri1 = sr[i * 3 + 1];
        const float ri2 = sr[i * 3 + 2];

        float hacc = 0.0f;  // half-wave partial of h[i, n]

        #pragma unroll
        for (int jt = 0; jt < 2; ++jt) {
            // A tile: 16 edge rows (j), K = component (3 valid + zero pad).
            const int jg = jt * 16 + lm;
            const int jc = (jg < NPT) ? jg : (NPT - 1);  // clamp; masked below
            const float e0 = ri0 - sr[jc * 3 + 0];
            const float e1 = ri1 - sr[jc * 3 + 1];
            const float e2 = ri2 - sr[jc * 3 + 2];
            v2f A;
            A.x = (hi == 0) ? e0 : e2;    // K = 0 or 2
            A.y = (hi == 0) ? e1 : 0.0f;  // K = 1 or 3 (pad)

            v8f C1, C2;
            #pragma unroll
            for (int v = 0; v < 8; ++v) { C1[v] = bias1; C2[v] = bias2; }

            // x = edges @ W1 + b1 ; y = edges @ W2 + b2  (16 j-rows x 16 features)
            v8f X = __builtin_amdgcn_wmma_f32_16x16x4_f32(
                false, A, false, B1t, (short)0, C1, false, false);
            v8f Y = __builtin_amdgcn_wmma_f32_16x16x4_f32(
                false, A, false, B2t, (short)0, C2, false, false);

            // silu(x)*silu(y), reduced over the j rows this lane holds (mask j>=20).
            #pragma unroll
            for (int v = 0; v < 8; ++v) {
                const int jrow = jt * 16 + v + 8 * hi;  // C/D row layout
                if (jrow < NPT) {
                    hacc += silu_f(X[v]) * silu_f(Y[v]);
                }
            }
        }

        // Fold the two lane halves (same column n lives in lanes L and L^16).
        hacc += __shfl_xor(hacc, 16, 32);
        // lanes 0-15 now hold h[i, n]; silu and accumulate (avoid double count).
        wsum += (hi == 0) ? silu_f(hacc) : 0.0f;
    }

    // Deterministic wave reduction.
    #pragma unroll
    for (int off = 16; off > 0; off >>= 1) wsum += __shfl_xor(wsum, off, 32);
    if (lane == 0) swave[wave] = wsum;
    __syncthreads();
    if (t == 0) {
        float s = 0.0f;
        #pragma unroll
        for (int w = 0; w < 8; ++w) s += swave[w];
        partial[b] = s;
    }
}

__global__ __launch_bounds__(32) void reduce_kernel(
    const float* __restrict__ partial, float* __restrict__ out, int n)
{
    const int lane = threadIdx.x;
    float s = 0.0f;
    for (int k = lane; k < n; k += 32) s += partial[k];
    #pragma unroll
    for (int off = 16; off > 0; off >>= 1) s += __shfl_xor(s, off, 32);
    if (lane == 0) out[0] = s;
}

extern "C" void kernel_launch(void* const* d_in, const int* in_sizes, int n_in,
                              void* d_out, int out_size, void* d_ws, size_t ws_size,
                              hipStream_t stream) {
    const float* r  = (const float*)d_in[0];
    const float* W1 = (const float*)d_in[1];
    const float* b1 = (const float*)d_in[2];
    const float* W2 = (const float*)d_in[3];
    const float* b2 = (const float*)d_in[4];
    float* out      = (float*)d_out;
    float* partial  = (float*)d_ws;   // B floats of scratch

    const int B = in_sizes[0] / (NPT * 3);  // 512

    edge_mlp_kernel<<<B, 256, 0, stream>>>(r, W1, b1, W2, b2, partial);
    reduce_kernel<<<1, 32, 0, stream>>>(partial, out, B);
}